// NeuralCircuitPolicies_9448928051476
// MI455X (gfx1250) — compile-verified
//
#include <hip/hip_runtime.h>

// ---------------------------------------------------------------------------
// Neural Circuit Policy (LTC) forward pass for MI455X (gfx1250, wave32).
//   Kernel 1: per-batch sequential ODE scan (latency-bound, LDS-resident state)
//   Kernel 2: output projection GEMM via V_WMMA_F32_16X16X4_F32
// ---------------------------------------------------------------------------

#define B_DIM      32
#define T_DIM      32
#define SENS       64
#define UNITS      224
#define MOTOR      32
#define OUT_LEN    32
#define UNFOLDS    6
#define EPS        1e-8f

#define NCHUNK     4                  // pre-dimension reduction chunks
#define BLK1       (UNITS * NCHUNK)   // 896 threads = 28 wave32s
#define PRE_PER_C  (UNITS / NCHUNK)   // 56 pre-neurons per chunk
#define SPRE_PER_C (SENS / NCHUNK)    // 16 sensory pre per chunk

__device__ __forceinline__ float sigmoidf(float x) {
    return 1.0f / (1.0f + __expf(-x));
}

// ---------------------------------------------------------------------------
// Kernel 1: one workgroup per batch element; v[224] lives in LDS.
// threads: tid = c*UNITS + j (chunk-major, post-minor); we use
// j = tid % UNITS (contiguous posts -> coalesced row reads), c = tid / UNITS.
// ---------------------------------------------------------------------------
__global__ __launch_bounds__(BLK1)
void ltc_scan_kernel(const float* __restrict__ inputs,       // [B,T,S]
                     const float* __restrict__ input_w,      // [S]
                     const float* __restrict__ input_b,      // [S]
                     const float* __restrict__ s_mu,         // [S,U]
                     const float* __restrict__ s_sigma,      // [S,U]
                     const float* __restrict__ s_w,          // [S,U]
                     const float* __restrict__ s_erev,       // [S,U]
                     const float* __restrict__ s_mask,       // [S,U]
                     const float* __restrict__ mu,           // [U,U]
                     const float* __restrict__ sigma,        // [U,U]
                     const float* __restrict__ w,            // [U,U]
                     const float* __restrict__ erev,         // [U,U]
                     const float* __restrict__ mask,         // [U,U]
                     const float* __restrict__ gleak,        // [U]
                     const float* __restrict__ vleak,        // [U]
                     const float* __restrict__ cm,           // [U]
                     const float* __restrict__ output_w,     // [M]
                     const float* __restrict__ output_b,     // [M]
                     float* __restrict__ motor_buf)          // [B*T, M]
{
    const int b   = blockIdx.x;
    const int tid = threadIdx.x;
    const int j   = tid % UNITS;   // post neuron (contiguous -> coalesced)
    const int c   = tid / UNITS;   // reduction chunk 0..3

    __shared__ float v_sh[UNITS];
    __shared__ float xt_sh[SENS];
    __shared__ float pnum[NCHUNK * UNITS];
    __shared__ float pden[NCHUNK * UNITS];

    if (tid < UNITS) v_sh[tid] = 0.0f;

    // per-post constants (used only by the c==0 combine lane)
    float cmtj = 0.0f, glj = 0.0f, glvlj = 0.0f;
    if (c == 0) {
        cmtj  = cm[j] * (float)UNFOLDS;   // cm / (dt/unfolds), dt = 1
        glj   = gleak[j];
        glvlj = glj * vleak[j];
    }
    __syncthreads();

    for (int t = 0; t < T_DIM; ++t) {
        // ---- affine input mapping into LDS ------------------------------
        if (tid < SENS) {
            const int base = (b * T_DIM + t) * SENS;
            xt_sh[tid] = inputs[base + tid] * input_w[tid] + input_b[tid];
            if (t + 1 < T_DIM)   // gfx1250 global_prefetch of next row
                __builtin_prefetch(&inputs[base + SENS + tid], 0, 1);
        }
        __syncthreads();

        // ---- sensory synapse sums (independent of v, once per step) -----
        float sn = 0.0f, sd = 0.0f;
        {
            const int s0 = c * SPRE_PER_C;
            #pragma unroll 4
            for (int s = s0; s < s0 + SPRE_PER_C; ++s) {
                const int idx = s * UNITS + j;
                const float act =
                    s_w[idx] * sigmoidf(s_sigma[idx] * (xt_sh[s] - s_mu[idx])) * s_mask[idx];
                sn += act * s_erev[idx];
                sd += act;
            }
        }
        pnum[c * UNITS + j] = sn;
        pden[c * UNITS + j] = sd;
        __syncthreads();

        float snum = 0.0f, sden = 0.0f;
        if (c == 0) {
            snum = pnum[j] + pnum[UNITS + j] + pnum[2 * UNITS + j] + pnum[3 * UNITS + j];
            sden = pden[j] + pden[UNITS + j] + pden[2 * UNITS + j] + pden[3 * UNITS + j];
        }
        __syncthreads();

        // ---- 6 fused ODE unfolds ---------------------------------------
        for (int k = 0; k < UNFOLDS; ++k) {
            float an = 0.0f, ad = 0.0f;
            const int i0 = c * PRE_PER_C;
            #pragma unroll 4
            for (int i = i0; i < i0 + PRE_PER_C; ++i) {
                const int idx = i * UNITS + j;
                const float act =
                    w[idx] * sigmoidf(sigma[idx] * (v_sh[i] - mu[idx])) * mask[idx];
                an += act * erev[idx];
                ad += act;
            }
            pnum[c * UNITS + j] = an;
            pden[c * UNITS + j] = ad;
            __syncthreads();   // all v_sh reads for this unfold are done

            if (c == 0) {
                const float num = cmtj * v_sh[j] + glvlj + snum
                                + pnum[j] + pnum[UNITS + j]
                                + pnum[2 * UNITS + j] + pnum[3 * UNITS + j];
                const float den = cmtj + glj + sden + EPS
                                + pden[j] + pden[UNITS + j]
                                + pden[2 * UNITS + j] + pden[3 * UNITS + j];
                v_sh[j] = num / den;
            }
            __syncthreads();
        }

        // ---- emit scaled motor trace for the WMMA projection ------------
        if (tid < MOTOR) {
            motor_buf[(b * T_DIM + t) * MOTOR + tid] =
                v_sh[tid] * output_w[tid] + output_b[tid];
        }
        __syncthreads();
    }
}

// ---------------------------------------------------------------------------
// Kernel 2: out[r, o] = motor[r, :] @ dense_w[:, o] + dense_b[o]
// r = b*T + t (1024 rows), K = 32, N = 32.  fp32 WMMA, one wave per 16x16 tile.
//
// V_WMMA_F32_16X16X4_F32 fragment layouts (ISA 7.12.2):
//   A (16x4 f32, 2 VGPRs): lane<16 -> K = k0,k0+1 ; lane>=16 -> K = k0+2,k0+3,
//                          M = lane&15 in both halves.
//   B (4x16 f32, 2 VGPRs): mirrored, N = lane&15, rows K as above.
//   C/D (16x16 f32, 8 VGPRs): VGPR r holds M = r (lanes 0-15) / M = r+8
//                          (lanes 16-31), N = lane&15.
// ---------------------------------------------------------------------------
typedef float v2f __attribute__((ext_vector_type(2)));
typedef float v8f __attribute__((ext_vector_type(8)));

__global__ __launch_bounds__(64)
void dense_wmma_kernel(const float* __restrict__ motor_buf,  // [1024, 32]
                       const float* __restrict__ dense_w,    // [32, 32] (K,N)
                       const float* __restrict__ dense_b,    // [32]
                       float* __restrict__ out)              // [1024, 32]
{
    const int lane = threadIdx.x;        // 0..31 (wave32)
    const int ct   = threadIdx.y;        // column tile 0..1
    const int rt   = blockIdx.x;         // row tile 0..63
    const int half = lane >> 4;          // 0 or 1
    const int l15  = lane & 15;

    const int row = rt * 16 + l15;       // A-fragment M
    const int col = ct * 16 + l15;       // B/D-fragment N

    v8f acc = {};
    #pragma unroll
    for (int k0 = 0; k0 < MOTOR; k0 += 4) {
        const int ka = k0 + half * 2;
        v2f a, bf;
        a.x  = motor_buf[row * MOTOR + ka];
        a.y  = motor_buf[row * MOTOR + ka + 1];
        bf.x = dense_w[ka * OUT_LEN + col];
        bf.y = dense_w[(ka + 1) * OUT_LEN + col];
        acc = __builtin_amdgcn_wmma_f32_16x16x4_f32(
                  /*neg_a=*/false, a, /*neg_b=*/false, bf,
                  /*c_mod=*/(short)0, acc,
                  /*reuse_a=*/false, /*reuse_b=*/false);
    }

    const float bias = dense_b[col];
    #pragma unroll
    for (int r = 0; r < 8; ++r) {
        const int orow = rt * 16 + r + half * 8;
        out[orow * OUT_LEN + col] = acc[r] + bias;
    }
}

// ---------------------------------------------------------------------------
extern "C" void kernel_launch(void* const* d_in, const int* in_sizes, int n_in,
                              void* d_out, int out_size, void* d_ws, size_t ws_size,
                              hipStream_t stream) {
    (void)in_sizes; (void)n_in; (void)out_size; (void)ws_size;

    const float* inputs    = (const float*)d_in[0];
    const float* input_w   = (const float*)d_in[1];
    const float* input_b   = (const float*)d_in[2];
    const float* s_mu      = (const float*)d_in[3];
    const float* s_sigma   = (const float*)d_in[4];
    const float* s_w       = (const float*)d_in[5];
    const float* s_erev    = (const float*)d_in[6];
    const float* s_mask    = (const float*)d_in[7];
    const float* mu        = (const float*)d_in[8];
    const float* sigma     = (const float*)d_in[9];
    const float* w         = (const float*)d_in[10];
    const float* erev      = (const float*)d_in[11];
    const float* mask      = (const float*)d_in[12];
    const float* gleak     = (const float*)d_in[13];
    const float* vleak     = (const float*)d_in[14];
    const float* cm        = (const float*)d_in[15];
    const float* output_w  = (const float*)d_in[16];
    const float* output_b  = (const float*)d_in[17];
    const float* dense_w   = (const float*)d_in[18];
    const float* dense_b   = (const float*)d_in[19];

    float* motor_buf = (float*)d_ws;             // [B*T, MOTOR] = 128 KB
    float* out       = (float*)d_out;            // [B, T, OUT_LEN]

    ltc_scan_kernel<<<dim3(B_DIM), dim3(BLK1), 0, stream>>>(
        inputs, input_w, input_b,
        s_mu, s_sigma, s_w, s_erev, s_mask,
        mu, sigma, w, erev, mask,
        gleak, vleak, cm, output_w, output_b, motor_buf);

    dense_wmma_kernel<<<dim3((B_DIM * T_DIM) / 16), dim3(32, 2), 0, stream>>>(
        motor_buf, dense_w, dense_b, out);
}